// SpectralConv_31293131719517
// MI455X (gfx1250) — compile-verified
//
#include <hip/hip_runtime.h>
#include <hip/hip_bf16.h>

typedef float v2f __attribute__((ext_vector_type(2)));
typedef float v8f __attribute__((ext_vector_type(8)));

#define PI2 6.283185307179586476925f

// D = A(16x4) * B(4x16) + C(16x16), fp32 WMMA on gfx1250
__device__ __forceinline__ v8f wmma4(v2f a, v2f b, v8f c) {
  return __builtin_amdgcn_wmma_f32_16x16x4_f32(false, a, false, b, (short)0, c,
                                               false, false);
}

__device__ __forceinline__ v8f vzero8() {
  v8f z = {0.f, 0.f, 0.f, 0.f, 0.f, 0.f, 0.f, 0.f};
  return z;
}

// ---------------- workspace layout (in floats) ----------------
// T1: fwd W-DFT   [2][256][32]   (cos | -sin)            16384
// T2: fwd H-DFT   [2][512][64]   (packed [Yr|Yi] K)      65536
// T3: inv H-DFT   [2][128][256]  (packed [Fr|Fi] K)      65536
// T4: inv W-irfft [64][256]      (norm+Hermitian folded) 16384
// Wpk: weights    [2048][2][64][64]                      16777216
// Y1: fwd W out   [1024][2][32][256]                     16777216  (aliased as Z)
// modes:          [2048][2][16][64]                      4194304
// outft:          [1024][2][64][32]                      4194304
static const size_t OFF_T1    = 0;
static const size_t OFF_T2    = 16384;
static const size_t OFF_T3    = 81920;
static const size_t OFF_T4    = 147456;
static const size_t OFF_WPK   = 163840;
static const size_t OFF_Y1    = 16941056;   // also Z after stage B
static const size_t OFF_MODES = 33718272;
static const size_t OFF_OUTFT = 37912576;
// total = 42106880 floats ~= 168.5 MB

// ---------------- table generation ----------------
__global__ void initTables(float* __restrict__ T1, float* __restrict__ T2,
                           float* __restrict__ T3, float* __restrict__ T4) {
  const int total = 16384 + 65536 + 65536 + 16384;
  for (int idx = blockIdx.x * blockDim.x + threadIdx.x; idx < total;
       idx += gridDim.x * blockDim.x) {
    if (idx < 16384) {
      // T1[s][k][n]: e^{-2pi i k n / 256} ; s=0 -> cos, s=1 -> -sin
      int t = idx, n = t & 31, k = (t >> 5) & 255, s = t >> 13;
      float th = PI2 * (float)((k * n) & 255) / 256.f;
      T1[t] = s ? -sinf(th) : cosf(th);
    } else if (idx < 16384 + 65536) {
      // T2[s][k][m]: fwd H DFT at 64 kept rows m0a; A packed [Yr(0:256)|Yi(256:512)]
      // mr = sum Yr*c + Yi*s ; mi = sum -Yr*s + Yi*c   (e^{-i th})
      int t = idx - 16384, m = t & 63, k = (t >> 6) & 511, s = t >> 15;
      int m0a = (m < 32) ? m : (192 + m);  // 224..255 for the bottom block
      int kk = k & 255;
      float th = PI2 * (float)((kk * m0a) & 255) / 256.f;
      float c = cosf(th), sn = sinf(th);
      T2[t] = (s == 0) ? ((k < 256) ? c : sn) : ((k < 256) ? -sn : c);
    } else if (idx < 16384 + 65536 + 65536) {
      // T3[s][k][h]: inverse H (e^{+i th}); A packed [Fr(0:64)|Fi(64:128)]
      // Zr = Fr*c - Fi*s ; Zi = Fr*s + Fi*c
      int t = idx - (16384 + 65536), h = t & 255, k = (t >> 8) & 127, s = t >> 15;
      int km = k & 63;
      int m0a = (km < 32) ? km : (192 + km);
      float th = PI2 * (float)((h * m0a) & 255) / 256.f;
      float c = cosf(th), sn = sinf(th);
      T3[t] = (s == 0) ? ((k < 64) ? c : -sn) : ((k < 64) ? sn : c);
    } else {
      // T4[k][w]: irfft along W with 1/(H*W) and Hermitian x2 folded in.
      // out[w] = sum_k<32 Zr[k]*a_k*cos + sum_k>=32 Zi[k-32]*(-a*sin)
      int t = idx - (16384 + 65536 + 65536), w = t & 255, k = t >> 8;
      int m1 = k & 31;
      float alpha = ((m1 == 0) ? 1.f : 2.f) / 65536.f;
      float th = PI2 * (float)((w * m1) & 255) / 256.f;
      T4[t] = (k < 32) ? alpha * cosf(th) : -alpha * sinf(th);
    }
  }
}

// ---------------- weight repack: [i][o][m0][m1] -> [mode][reim][i][o] ----------------
__global__ void repackW(const float* __restrict__ w0r, const float* __restrict__ w0i,
                        const float* __restrict__ w1r, const float* __restrict__ w1i,
                        float* __restrict__ Wpk) {
  const size_t N = (size_t)2048 * 2 * 64 * 64;
  for (size_t idx = (size_t)blockIdx.x * blockDim.x + threadIdx.x; idx < N;
       idx += (size_t)gridDim.x * blockDim.x) {
    const int o = (int)(idx & 63);
    const int i = (int)((idx >> 6) & 63);
    const int re = (int)((idx >> 12) & 1);
    const int mode = (int)(idx >> 13);
    const int m = mode >> 5, m1 = mode & 31;
    const float* src = (m < 32) ? (re ? w0i : w0r) : (re ? w1i : w1r);
    const int m0 = m & 31;
    Wpk[idx] = src[(((size_t)i * 64 + o) * 32 + m0) * 32 + m1];
  }
}

// ---------------- stage A: truncated rfft along W ----------------
// rows = (b,i,h) [1M], K=256 (w), N=32 (m1); out Y1[bi][reim][m1][h]
__global__ __launch_bounds__(32) void stageA(const float* __restrict__ x,
                                             const float* __restrict__ T1,
                                             float* __restrict__ Y1) {
  __shared__ float As[16 * 257];
  const int lane = threadIdx.x;
  const int tile = blockIdx.x;  // 65536
  const int bi = tile >> 4;
  const int h0 = (tile & 15) << 4;
  const float* xrow = x + (size_t)bi * (256u * 256u) + (size_t)h0 * 256u;
  for (int i = lane; i < 16 * 256; i += 32)
    As[(i >> 8) * 257 + (i & 255)] = xrow[i];
  __syncthreads();
  const float* T1c = T1;
  const float* T1s = T1 + 256 * 32;
  const int half = lane >> 4, lm = lane & 15;
  v8f acc[4];
  for (int q = 0; q < 4; ++q) acc[q] = vzero8();
  for (int k0 = 0; k0 < 256; k0 += 4) {
    const int kk = k0 + 2 * half;
    v2f a = {As[lm * 257 + kk], As[lm * 257 + kk + 1]};
#pragma unroll
    for (int nt = 0; nt < 2; ++nt) {
      const int n = nt * 16 + lm;
      v2f bc = {T1c[kk * 32 + n], T1c[(kk + 1) * 32 + n]};
      v2f bs = {T1s[kk * 32 + n], T1s[(kk + 1) * 32 + n]};
      acc[nt * 2 + 0] = wmma4(a, bc, acc[nt * 2 + 0]);
      acc[nt * 2 + 1] = wmma4(a, bs, acc[nt * 2 + 1]);
    }
  }
  float* Yb = Y1 + (size_t)bi * (2 * 32 * 256);
#pragma unroll
  for (int nt = 0; nt < 2; ++nt)
#pragma unroll
    for (int re = 0; re < 2; ++re)
#pragma unroll
      for (int j = 0; j < 8; ++j) {
        const int m1 = nt * 16 + lm;
        const int hh = h0 + j + 8 * half;
        Yb[(re * 32 + m1) * 256 + hh] = acc[nt * 2 + re][j];
      }
}

// ---------------- stage B: fft along H at 64 kept rows ----------------
// rows = (bi,m1) [32768], K=512 ([Yr|Yi] over h), N=64 (m); out modes[mode][reim][bi]
__global__ __launch_bounds__(32) void stageB(const float* __restrict__ Y1,
                                             const float* __restrict__ T2,
                                             float* __restrict__ modes) {
  __shared__ float As[16 * 513];
  const int lane = threadIdx.x;
  const int tile = blockIdx.x;  // 2048
  const int bi = tile >> 1;
  const int m1b = (tile & 1) << 4;
  const float* Yb = Y1 + (size_t)bi * (2 * 32 * 256);
  for (int i = lane; i < 16 * 512; i += 32) {
    const int r = i >> 9, k = i & 511;
    As[r * 513 + k] = Yb[((k >> 8) * 32 + (m1b + r)) * 256 + (k & 255)];
  }
  __syncthreads();
  const float* T2a = T2;
  const float* T2b = T2 + 512 * 64;
  const int half = lane >> 4, lm = lane & 15;
  v8f accR[4], accI[4];
  for (int q = 0; q < 4; ++q) { accR[q] = vzero8(); accI[q] = vzero8(); }
  for (int k0 = 0; k0 < 512; k0 += 4) {
    const int kk = k0 + 2 * half;
    v2f a = {As[lm * 513 + kk], As[lm * 513 + kk + 1]};
#pragma unroll
    for (int nt = 0; nt < 4; ++nt) {
      const int n = nt * 16 + lm;
      v2f ba = {T2a[kk * 64 + n], T2a[(kk + 1) * 64 + n]};
      v2f bb = {T2b[kk * 64 + n], T2b[(kk + 1) * 64 + n]};
      accR[nt] = wmma4(a, ba, accR[nt]);
      accI[nt] = wmma4(a, bb, accI[nt]);
    }
  }
#pragma unroll
  for (int nt = 0; nt < 4; ++nt)
#pragma unroll
    for (int j = 0; j < 8; ++j) {
      const int m = nt * 16 + lm;
      const int m1 = m1b + j + 8 * half;
      float* p = modes + (size_t)(m * 32 + m1) * 2048 + bi;
      p[0] = accR[nt][j];
      p[1024] = accI[nt][j];
    }
}

// ---------------- stage C: per-mode complex channel mix ----------------
// per mode: [16(b) x 64(i)] * [64(i) x 64(o)] complex; out outft[bo][reim][m][m1]
__global__ __launch_bounds__(32) void stageC(const float* __restrict__ modes,
                                             const float* __restrict__ Wpk,
                                             float* __restrict__ outft) {
  __shared__ float As[2 * 16 * 65];
  const int lane = threadIdx.x;
  const int mode = blockIdx.x;  // 2048
  const int m = mode >> 5, m1 = mode & 31;
  const float* Mb = modes + (size_t)mode * 2048;
  for (int i = lane; i < 2048; i += 32) {
    const int re = i >> 10, b = (i >> 6) & 15, ii = i & 63;
    As[(re * 16 + b) * 65 + ii] = Mb[i];
  }
  __syncthreads();
  const float* Wr = Wpk + (size_t)mode * 8192;
  const float* Wi = Wr + 4096;
  const float* ARr = As;
  const float* ARi = As + 16 * 65;
  const int half = lane >> 4, lm = lane & 15;
  v8f Dr[4], Di[4];
  for (int q = 0; q < 4; ++q) { Dr[q] = vzero8(); Di[q] = vzero8(); }
  for (int k0 = 0; k0 < 64; k0 += 4) {
    const int kk = k0 + 2 * half;
    v2f ar = {ARr[lm * 65 + kk], ARr[lm * 65 + kk + 1]};
    v2f ai = {ARi[lm * 65 + kk], ARi[lm * 65 + kk + 1]};
    v2f nai = {-ai.x, -ai.y};  // f32 WMMA has no A-neg modifier; negate in VALU
#pragma unroll
    for (int nt = 0; nt < 4; ++nt) {
      const int n = nt * 16 + lm;
      v2f br = {Wr[kk * 64 + n], Wr[(kk + 1) * 64 + n]};
      v2f bI = {Wi[kk * 64 + n], Wi[(kk + 1) * 64 + n]};
      Dr[nt] = wmma4(ar, br, Dr[nt]);
      Dr[nt] = wmma4(nai, bI, Dr[nt]);
      Di[nt] = wmma4(ar, bI, Di[nt]);
      Di[nt] = wmma4(ai, br, Di[nt]);
    }
  }
#pragma unroll
  for (int nt = 0; nt < 4; ++nt)
#pragma unroll
    for (int j = 0; j < 8; ++j) {
      const int b = j + 8 * half;
      const int o = nt * 16 + lm;
      float* p = outft + (size_t)(b * 64 + o) * 4096 + (size_t)m * 32 + m1;
      p[0] = Dr[nt][j];
      p[2048] = Di[nt][j];
    }
}

// ---------------- stage D1: inverse fft along H ----------------
// rows = (bo,m1) [32768], K=128 ([Fr|Fi] over m), N=256 (h, split 4 ways)
__global__ __launch_bounds__(32) void stageD1(const float* __restrict__ outft,
                                              const float* __restrict__ T3,
                                              float* __restrict__ Z) {
  __shared__ float As[16 * 129];
  const int lane = threadIdx.x;
  const int blk = blockIdx.x;  // 8192
  const int hg = blk & 3;
  const int tile = blk >> 2;
  const int bo = tile >> 1;
  const int m1b = (tile & 1) << 4;
  const float* OF = outft + (size_t)bo * 4096;
  for (int i = lane; i < 16 * 128; i += 32) {
    const int r = i & 15, k = i >> 4;
    As[r * 129 + k] = OF[((k >> 6) * 64 + (k & 63)) * 32 + m1b + r];
  }
  __syncthreads();
  const float* T3a = T3;
  const float* T3b = T3 + 128 * 256;
  const int half = lane >> 4, lm = lane & 15;
  v8f Zr[4], Zi[4];
  for (int q = 0; q < 4; ++q) { Zr[q] = vzero8(); Zi[q] = vzero8(); }
  for (int k0 = 0; k0 < 128; k0 += 4) {
    const int kk = k0 + 2 * half;
    v2f a = {As[lm * 129 + kk], As[lm * 129 + kk + 1]};
#pragma unroll
    for (int nt = 0; nt < 4; ++nt) {
      const int h = hg * 64 + nt * 16 + lm;
      v2f ba = {T3a[kk * 256 + h], T3a[(kk + 1) * 256 + h]};
      v2f bb = {T3b[kk * 256 + h], T3b[(kk + 1) * 256 + h]};
      Zr[nt] = wmma4(a, ba, Zr[nt]);
      Zi[nt] = wmma4(a, bb, Zi[nt]);
    }
  }
#pragma unroll
  for (int nt = 0; nt < 4; ++nt)
#pragma unroll
    for (int j = 0; j < 8; ++j) {
      const int m1 = m1b + j + 8 * half;
      const int h = hg * 64 + nt * 16 + lm;
      float* p = Z + ((size_t)bo * 256 + h) * 64 + m1;
      p[0] = Zr[nt][j];
      p[32] = Zi[nt][j];
    }
}

// ---------------- stage D2: irfft along W ----------------
// rows = (bo,h) [256K... per bo 256], K=64 ([Zr|Zi] over m1), N=256 (w)
// 4 waves/block share one LDS A-tile; wave v covers w in [v*64, v*64+64)
__global__ __launch_bounds__(128) void stageD2(const float* __restrict__ Z,
                                               const float* __restrict__ T4,
                                               float* __restrict__ out) {
  __shared__ float Zs[16 * 65];
  const int tid = threadIdx.x;
  const int lane = tid & 31;
  const int wave = tid >> 5;
  const int blk = blockIdx.x;  // 16384
  const int bo = blk >> 4;
  const int ht = blk & 15;
  const float* src = Z + ((size_t)bo * 256 + (size_t)ht * 16) * 64;
  for (int i = tid; i < 1024; i += 128)
    Zs[(i >> 6) * 65 + (i & 63)] = src[i];
  __syncthreads();
  const int half = lane >> 4, lm = lane & 15;
  const int wbase = wave * 64;
  v8f acc[4];
  for (int q = 0; q < 4; ++q) acc[q] = vzero8();
  for (int k0 = 0; k0 < 64; k0 += 4) {
    const int kk = k0 + 2 * half;
    v2f a = {Zs[lm * 65 + kk], Zs[lm * 65 + kk + 1]};
#pragma unroll
    for (int nt = 0; nt < 4; ++nt) {
      const int w = wbase + nt * 16 + lm;
      v2f b = {T4[kk * 256 + w], T4[(kk + 1) * 256 + w]};
      acc[nt] = wmma4(a, b, acc[nt]);
    }
  }
#pragma unroll
  for (int nt = 0; nt < 4; ++nt)
#pragma unroll
    for (int j = 0; j < 8; ++j) {
      const int h = ht * 16 + j + 8 * half;
      const int w = wbase + nt * 16 + lm;
      out[((size_t)bo * 256 + h) * 256 + w] = acc[nt][j];
    }
}

extern "C" void kernel_launch(void* const* d_in, const int* in_sizes, int n_in,
                              void* d_out, int out_size, void* d_ws, size_t ws_size,
                              hipStream_t stream) {
  const float* x   = (const float*)d_in[0];
  const float* w0r = (const float*)d_in[1];
  const float* w0i = (const float*)d_in[2];
  const float* w1r = (const float*)d_in[3];
  const float* w1i = (const float*)d_in[4];
  float* ws  = (float*)d_ws;
  float* out = (float*)d_out;

  float* T1    = ws + OFF_T1;
  float* T2    = ws + OFF_T2;
  float* T3    = ws + OFF_T3;
  float* T4    = ws + OFF_T4;
  float* Wpk   = ws + OFF_WPK;
  float* Y1    = ws + OFF_Y1;     // stage A/B buffer
  float* Zbuf  = ws + OFF_Y1;     // Y1 dead after stage B -> reuse as Z
  float* mbuf  = ws + OFF_MODES;
  float* ofbuf = ws + OFF_OUTFT;

  initTables<<<160, 256, 0, stream>>>(T1, T2, T3, T4);
  repackW<<<4096, 256, 0, stream>>>(w0r, w0i, w1r, w1i, Wpk);
  stageA<<<65536, 32, 0, stream>>>(x, T1, Y1);
  stageB<<<2048, 32, 0, stream>>>(Y1, T2, mbuf);
  stageC<<<2048, 32, 0, stream>>>(mbuf, Wpk, ofbuf);
  stageD1<<<8192, 32, 0, stream>>>(ofbuf, T3, Zbuf);
  stageD2<<<16384, 128, 0, stream>>>(Zbuf, T4, out);
}